// SafetyLayer_83159156785239
// MI455X (gfx1250) — compile-verified
//
#include <hip/hip_runtime.h>
#include <math.h>

typedef __attribute__((ext_vector_type(2))) float v2f;
typedef __attribute__((ext_vector_type(8))) float v8f;

#define STEP_SIZE 0.1f
#define D_LIMIT   0.05f
#define EPSV      1e-9f

__device__ __forceinline__ float clamp01(float x) {
    return fminf(fmaxf(x, 0.0f), 1.0f);
}

// One wave32 per batch element.
// lane = robot link for FK stage; lane%16 = patient capsule for distance stage.
__global__ __launch_bounds__(256)
void SafetyLayer_kernel(const float* __restrict__ joint_state,
                        const float* __restrict__ action_nominal,
                        const float* __restrict__ link_lengths,
                        const float* __restrict__ robot_radii,
                        const float* __restrict__ p_start,
                        const float* __restrict__ p_end,
                        const float* __restrict__ p_radii,
                        const float* __restrict__ log_lambda,
                        float* __restrict__ out_action,
                        float* __restrict__ out_penalty,
                        float* __restrict__ out_mindist,
                        int B) {
    const int lane = threadIdx.x & 31;
    const int wave = blockIdx.x * (blockDim.x >> 5) + (threadIdx.x >> 5);
    if (wave >= B) return;            // wave-uniform: EXEC stays all-ones for WMMA
    const int b = wave;

    // ---------------- Forward kinematics (wave prefix scans) ----------------
    float q  = joint_state[b * 32 + lane];
    float an = action_nominal[b * 32 + lane];
    out_action[b * 32 + lane] = an;                 // passthrough output 0

    float theta = fmaf(an, STEP_SIZE, q);
    #pragma unroll
    for (int d = 1; d < 32; d <<= 1) {              // inclusive scan of angles
        float t = __shfl(theta, lane - d, 32);
        if (lane >= d) theta += t;
    }

    float L  = link_lengths[lane];
    float dx = L * cosf(theta);                     // link direction * length (d1), z = 0
    float dy = L * sinf(theta);

    float ex = dx, ey = dy;                         // inclusive scan -> link end points
    #pragma unroll
    for (int d = 1; d < 32; d <<= 1) {
        float tx = __shfl(ex, lane - d, 32);
        float ty = __shfl(ey, lane - d, 32);
        if (lane >= d) { ex += tx; ey += ty; }
    }
    float rsx = ex - dx;                            // r_start (exclusive scan), z = 0
    float rsy = ey - dy;
    float rrad = robot_radii[lane];

    // ---------------- Patient capsule (per-lane, p = lane % 16) ----------------
    const int p = lane & 15;
    const float* ps = p_start + (b * 16 + p) * 3;
    const float* pe = p_end   + (b * 16 + p) * 3;
    float psx = ps[0], psy = ps[1], psz = ps[2];
    float d2x = pe[0] - psx, d2y = pe[1] - psy, d2z = pe[2] - psz;
    float e_n  = d2x * d2x + d2y * d2y + d2z * d2z;       // d2 . d2
    float t2_n = d2x * psx + d2y * psy + d2z * psz;       // d2 . p_start
    float prad = p_radii[b * 16 + p];

    // Hoisted per-capsule reciprocal (loop-invariant per lane)
    const bool  eok  = (e_n > EPSV);
    const float inve = __builtin_amdgcn_rcpf(fmaxf(e_n, EPSV));

    const bool lowhalf = (lane < 16);

    // B matrices (4x16, K rows striped over lanes; rows K=2,3 = 0 pad)
    v2f Bm1;  // d2_xy
    Bm1[0] = lowhalf ? d2x : 0.0f;
    Bm1[1] = lowhalf ? d2y : 0.0f;
    v2f Bm2;  // p_start_xy
    Bm2[0] = lowhalf ? psx : 0.0f;
    Bm2[1] = lowhalf ? psy : 0.0f;

    float minv = 1e30f;
    const int hi8 = (lane >> 4) << 3;   // 8 * (lane >= 16)

    #pragma unroll
    for (int t = 0; t < 2; ++t) {       // two M-tiles of 16 links each
        // A matrices (16x4, row m split across lane m / lane m+16; K=2,3 = 0 pad)
        const int msrc = 16 * t + (lane & 15);
        float sdx  = __shfl(dx,  msrc, 32);
        float sdy  = __shfl(dy,  msrc, 32);
        float srsx = __shfl(rsx, msrc, 32);
        float srsy = __shfl(rsy, msrc, 32);
        v2f A1;  // d1_xy
        A1[0] = lowhalf ? sdx  : 0.0f;
        A1[1] = lowhalf ? sdy  : 0.0f;
        v2f A2;  // r_start_xy
        A2[0] = lowhalf ? srsx : 0.0f;
        A2[1] = lowhalf ? srsy : 0.0f;

        v8f Z = {};
        // b[m,n]  = d1 . d2
        v8f Db  = __builtin_amdgcn_wmma_f32_16x16x4_f32(false, A1, false, Bm1,
                                                        (short)0, Z, false, false);
        // c2[m,n] = d1 . p_start  (c = d1.r_start - c2)
        v8f Dc2 = __builtin_amdgcn_wmma_f32_16x16x4_f32(false, A1, false, Bm2,
                                                        (short)0, Z, false, false);
        // f1[m,n] = r_start . d2  (f = f1 - d2.p_start)
        v8f Df1 = __builtin_amdgcn_wmma_f32_16x16x4_f32(false, A2, false, Bm1,
                                                        (short)0, Z, false, false);

        #pragma unroll
        for (int v = 0; v < 8; ++v) {
            // D layout: VGPR v, lane l -> (m = v + 8*(l>=16), n = l%16)
            const int ms = 16 * t + v + hi8;        // global link index for this element
            float dxm  = __shfl(dx,   ms, 32);
            float dym  = __shfl(dy,   ms, 32);
            float rsxm = __shfl(rsx,  ms, 32);
            float rsym = __shfl(rsy,  ms, 32);
            float rrm  = __shfl(rrad, ms, 32);

            float am = fmaf(dxm, dxm, dym * dym);               // d1 . d1
            float bb = Db[v];
            float cc = fmaf(dxm, rsxm, dym * rsym) - Dc2[v];    // d1 . (r_start - p_start)
            float ff = Df1[v] - t2_n;                           // d2 . (r_start - p_start)

            // s = denom>EPS ? clip((b*f - c*e)/max(denom,EPS)) : 0   (branchless, v_rcp)
            float denom = fmaf(am, e_n, -(bb * bb));
            float invd  = __builtin_amdgcn_rcpf(fmaxf(denom, EPSV));
            float s0    = clamp01(fmaf(bb, ff, -(cc * e_n)) * invd);
            float s     = (denom > EPSV) ? s0 : 0.0f;

            // t = e>EPS ? (b*s + f)/max(e,EPS) : 0
            float tt = eok ? (fmaf(bb, s, ff) * inve) : 0.0f;
            float tc = clamp01(tt);

            // s = (t != clip(t)) ? (a>EPS ? clip((b*tc - c)/max(a,EPS)) : 0) : s
            float inva = __builtin_amdgcn_rcpf(fmaxf(am, EPSV));
            float sa0  = clamp01(fmaf(bb, tc, -cc) * inva);
            float sa   = (am > EPSV) ? sa0 : 0.0f;
            s = (tt != tc) ? sa : s;

            float c1x = fmaf(dxm, s, rsxm);
            float c1y = fmaf(dym, s, rsym);
            float ux = c1x - fmaf(d2x, tc, psx);
            float uy = c1y - fmaf(d2y, tc, psy);
            float uz = -fmaf(d2z, tc, psz);                     // c1z == 0
            float dist = sqrtf(fmaf(ux, ux, fmaf(uy, uy, fmaf(uz, uz, 1e-12f))));
            minv = fminf(minv, dist - (rrm + prad));
        }
    }

    // ---------------- wave min-reduction + tail outputs ----------------
    #pragma unroll
    for (int off = 16; off; off >>= 1)
        minv = fminf(minv, __shfl_xor(minv, off, 32));

    if (lane == 0) {
        float md  = minv;
        float lam = expf(log_lambda[0]);
        float mag = lam * (D_LIMIT - md);
        out_penalty[b] = (md < D_LIMIT) ? mag : 0.0f;
        out_mindist[b] = md;
    }
}

extern "C" void kernel_launch(void* const* d_in, const int* in_sizes, int n_in,
                              void* d_out, int out_size, void* d_ws, size_t ws_size,
                              hipStream_t stream) {
    (void)n_in; (void)d_ws; (void)ws_size; (void)out_size;
    const float* joint_state    = (const float*)d_in[0];
    const float* action_nominal = (const float*)d_in[1];
    const float* link_lengths   = (const float*)d_in[2];
    const float* robot_radii    = (const float*)d_in[3];
    const float* p_start        = (const float*)d_in[4];
    const float* p_end          = (const float*)d_in[5];
    const float* p_radii        = (const float*)d_in[6];
    const float* log_lambda     = (const float*)d_in[7];

    const int B = in_sizes[0] / 32;          // [B, R=32]
    float* out_action  = (float*)d_out;       // [B*32]
    float* out_penalty = out_action + B * 32; // [B]
    float* out_mindist = out_penalty + B;     // [B]

    const int threads = 256;                  // 8 waves/block
    const int wavesPerBlock = threads / 32;
    const int blocks = (B + wavesPerBlock - 1) / wavesPerBlock;
    SafetyLayer_kernel<<<blocks, threads, 0, stream>>>(
        joint_state, action_nominal, link_lengths, robot_radii,
        p_start, p_end, p_radii, log_lambda,
        out_action, out_penalty, out_mindist, B);
}